// NMSLoss2_87136296501785
// MI455X (gfx1250) — compile-verified
//
#include <hip/hip_runtime.h>
#include <hip/hip_bf16.h>
#include <math.h>

// Problem shape (fixed by the harness' setup_inputs): B=8, N=2048, G=64.
#define B_BATCH   8
#define N_PROP    2048
#define N_GT      64
#define NTHREADS  256
#define NWAVES    (NTHREADS / 32)
#define EPS_F     1e-6f
#define NMS_THR_F 0.5f

// ---- wave32 helpers -------------------------------------------------------

__device__ __forceinline__ unsigned long long shfl_xor_u64(unsigned long long v, int m) {
  unsigned lo = __shfl_xor((unsigned)v, m);
  unsigned hi = __shfl_xor((unsigned)(v >> 32), m);
  return ((unsigned long long)hi << 32) | lo;
}

__device__ __forceinline__ float iou_boxes(float ax1, float ay1, float ax2, float ay2,
                                           float bx1, float by1, float bx2, float by2) {
  float aarea = (ax2 - ax1) * (ay2 - ay1);
  float barea = (bx2 - bx1) * (by2 - by1);
  float lx = fmaxf(ax1, bx1), ly = fmaxf(ay1, by1);
  float rx = fminf(ax2, bx2), ry = fminf(ay2, by2);
  float w = fmaxf(rx - lx, 0.0f), h = fmaxf(ry - ly, 0.0f);
  float inter = w * h;
  return inter / (aarea + barea - inter + EPS_F);
}

// Low 32 bits of a generic pointer to a __shared__ object == LDS byte offset
// (LDS aperture is 4GB-aligned; async-DMA VDST takes wave-relative LDS bytes).
__device__ __forceinline__ unsigned lds_off(const void* p) {
  return (unsigned)(unsigned long long)p;
}

__device__ __forceinline__ void async_copy_b128(unsigned loff, const void* gaddr) {
  // CDNA5 async global->LDS DMA (tracked with ASYNCcnt), 16B per lane.
  asm volatile("global_load_async_to_lds_b128 %0, %1, off"
               :: "v"(loff), "v"(gaddr) : "memory");
}

// ---- main kernel: one workgroup per batch ---------------------------------

__global__ __launch_bounds__(NTHREADS)
void nms_loss_batch_kernel(const int* __restrict__ gti_g,     // (B,N)
                           const float* __restrict__ gtb_g,   // (B,G,4)
                           const float* __restrict__ prop_g,  // (B,N,5)
                           float* __restrict__ out_bp) {      // (B,2): push,pull
  __shared__ __align__(16) float s_prop[N_PROP * 5];  // AoS x1,y1,x2,y2,s (stride 5: bank-conflict free)
  __shared__ __align__(16) float s_per[N_PROP];       // -log(EPS+gpi_per)*score
  __shared__ __align__(16) int   s_gti[N_PROP];
  __shared__ unsigned char       s_act[N_PROP];
  __shared__ int                 s_rec[N_GT];
  __shared__ unsigned long long  s_red64[NWAVES];
  __shared__ float               s_redf[NWAVES];
  __shared__ int                 s_redi[3 * NWAVES];
  __shared__ int                 s_bcast[3];          // i, i_gt(raw), remCount
  __shared__ int                 s_nactive;
  __shared__ int                 s_npos;

  const int b    = blockIdx.x;
  const int tid  = threadIdx.x;
  const int lane = tid & 31;
  const int wv   = tid >> 5;

  const float* gprop = prop_g + (size_t)b * N_PROP * 5;
  const int*   ggti  = gti_g  + (size_t)b * N_PROP;
  const float* ggtb  = gtb_g  + (size_t)b * N_GT * 4;

  // ---- Phase 0: async DMA stage global -> LDS (proposals 40KB, gti 8KB) ----
  for (int c = tid; c < (N_PROP * 5) / 4; c += NTHREADS)
    async_copy_b128(lds_off(&s_prop[4 * c]), (const void*)(gprop + 4 * c));
  for (int c = tid; c < N_PROP / 4; c += NTHREADS)
    async_copy_b128(lds_off(&s_gti[4 * c]), (const void*)(ggti + 4 * c));
  asm volatile("s_wait_asynccnt 0" ::: "memory");
  __syncthreads();

  // ---- Phase 1: per-proposal precompute ----
  int npos_local = 0;
  for (int k = tid; k < N_PROP; k += NTHREADS) {
    float x1 = s_prop[5 * k + 0], y1 = s_prop[5 * k + 1];
    float x2 = s_prop[5 * k + 2], y2 = s_prop[5 * k + 3];
    float sc = s_prop[5 * k + 4];
    int g = s_gti[k];
    int gw = g;
    if (gw < 0) gw += N_GT;                 // python-style negative wrap
    if (gw < 0) gw = 0;
    if (gw >= N_GT) gw = N_GT - 1;          // JAX clamps OOB gathers
    float gx1 = ggtb[4 * gw + 0], gy1 = ggtb[4 * gw + 1];
    float gx2 = ggtb[4 * gw + 2], gy2 = ggtb[4 * gw + 3];
    float gpi = iou_boxes(gx1, gy1, gx2, gy2, x1, y1, x2, y2);
    s_per[k] = -logf(EPS_F + gpi) * sc;
    s_act[k] = (g >= 0) ? (unsigned char)1 : (unsigned char)0;
    if (g >= 0) npos_local++;
  }
  if (tid < N_GT) s_rec[tid] = -1;
  for (int off = 16; off; off >>= 1) npos_local += __shfl_xor(npos_local, off);
  if (lane == 0) s_redi[wv] = npos_local;
  __syncthreads();
  if (tid == 0) {
    int np = 0;
    for (int w = 0; w < NWAVES; ++w) np += s_redi[w];
    s_npos = np;
    s_nactive = np;
  }
  __syncthreads();

  // ---- Phase 2: sequential NMS scan (thread 0 holds scalar accumulators) ----
  float pull_sum = 0.0f, push_sum = 0.0f, pull_cnt = 0.0f, push_cnt = 0.0f;

  for (int it = 0; it < N_PROP; ++it) {
    if (s_nactive <= 0) break;  // remaining reference steps are exact no-ops

    // (a) argmax over active scores; tie-break -> lowest index.
    unsigned long long key = 0ULL;
    for (int k = tid; k < N_PROP; k += NTHREADS) {
      if (s_act[k]) {
        unsigned sb = __float_as_uint(s_prop[5 * k + 4]) + 1u;  // scores >= 0
        unsigned long long kk =
            ((unsigned long long)sb << 32) | (unsigned)(N_PROP - 1 - k);
        if (kk > key) key = kk;
      }
    }
    for (int off = 16; off; off >>= 1) {
      unsigned long long o = shfl_xor_u64(key, off);
      if (o > key) key = o;
    }
    if (lane == 0) s_red64[wv] = key;
    __syncthreads();

    // (b) scalar step on thread 0: select i, pull bookkeeping, rec update.
    if (tid == 0) {
      unsigned long long best = 0ULL;
      for (int w = 0; w < NWAVES; ++w)
        if (s_red64[w] > best) best = s_red64[w];
      int i  = N_PROP - 1 - (int)(best & 0xffffffffu);
      int ig = s_gti[i];                       // selected among active => >= 0
      int igw = ig;
      if (igw < 0) igw += N_GT;
      if (igw < 0) igw = 0;
      if (igw >= N_GT) igw = N_GT - 1;
      float score_i = s_prop[5 * i + 4];
      s_act[i] = 0;                            // rem = active \ {i}
      int remc = s_nactive - 1;

      int r = s_rec[igw];
      bool seen = (r >= 0);
      int max_idx = seen ? r : 0;              // max(rec, 0)
      float siou = iou_boxes(s_prop[5 * max_idx + 0], s_prop[5 * max_idx + 1],
                             s_prop[5 * max_idx + 2], s_prop[5 * max_idx + 3],
                             s_prop[5 * i + 0], s_prop[5 * i + 1],
                             s_prop[5 * i + 2], s_prop[5 * i + 3]);
      float max_s_iou = fmaxf(siou, EPS_F);
      float pull = -logf(1.0f - NMS_THR_F + max_s_iou) * score_i;
      if (seen && remc > 0) pull_sum += pull;  // needs has_rem
      if (seen)             pull_cnt += 1.0f;  // does NOT need has_rem
      if (!seen) s_rec[igw] = i;               // rec updated BEFORE seen_s use

      s_bcast[0] = i;
      s_bcast[1] = ig;                         // raw value for gti!=i_gt test
      s_bcast[2] = remc;
    }
    __syncthreads();

    // (c) parallel overlap/push pass over remaining proposals.
    const int i    = s_bcast[0];
    const int igr  = s_bcast[1];
    const int remc = s_bcast[2];
    const float ix1 = s_prop[5 * i + 0], iy1 = s_prop[5 * i + 1];
    const float ix2 = s_prop[5 * i + 2], iy2 = s_prop[5 * i + 3];

    int pm = 0, qc = 0, rm = 0;
    float qs = 0.0f;
    for (int k = tid; k < N_PROP; k += NTHREADS) {
      if (!s_act[k]) continue;
      float v = iou_boxes(ix1, iy1, ix2, iy2,
                          s_prop[5 * k + 0], s_prop[5 * k + 1],
                          s_prop[5 * k + 2], s_prop[5 * k + 3]);
      if (v > NMS_THR_F) {                     // overlap = rem & iou>thr
        rm++;
        s_act[k] = 0;                          // active_next = rem & ~overlap
        int gk = s_gti[k];
        if (gk != igr) {                       // push_mask
          pm++;
          int rk = gk;
          if (rk < 0) rk += N_GT;
          if (rk < 0) rk = 0;
          if (rk >= N_GT) rk = N_GT - 1;
          if (s_rec[rk] < 0) {                 // q = push_mask & ~seen_s
            qc++;
            qs += s_per[k];
          }
        }
      }
    }
    for (int off = 16; off; off >>= 1) {
      pm += __shfl_xor(pm, off);
      qc += __shfl_xor(qc, off);
      rm += __shfl_xor(rm, off);
      qs += __shfl_xor(qs, off);
    }
    if (lane == 0) {
      s_redi[wv]              = pm;
      s_redi[NWAVES + wv]     = qc;
      s_redi[2 * NWAVES + wv] = rm;
      s_redf[wv]              = qs;
    }
    __syncthreads();

    if (tid == 0) {
      int PM = 0, QC = 0, RM = 0;
      float QS = 0.0f;
      for (int w = 0; w < NWAVES; ++w) {
        PM += s_redi[w];
        QC += s_redi[NWAVES + w];
        RM += s_redi[2 * NWAVES + w];
        QS += s_redf[w];
      }
      // has_push tests push_mask (PM); divisor counts q (QC).
      if (remc > 0 && PM > 0) {
        push_sum += QS / ((float)QC + EPS_F);
        push_cnt += 1.0f;
      }
      s_nactive = s_nactive - 1 - RM;
    }
    __syncthreads();
  }

  if (tid == 0) {
    float gate  = (s_npos > 1) ? 1.0f : 0.0f;
    float pushb = gate * push_sum / (push_cnt + EPS_F);
    float pullb = gate * pull_sum / (pull_cnt + EPS_F);
    out_bp[2 * b + 0] = pushb;
    out_bp[2 * b + 1] = pullb;
  }
}

// ---- finalize: mean over batches, stack [push*PUSH_W, pull*PULL_W] --------

__global__ void nms_loss_finalize_kernel(const float* __restrict__ bp,
                                         float* __restrict__ out) {
  if (threadIdx.x == 0 && blockIdx.x == 0) {
    float ps = 0.0f, pl = 0.0f;
    for (int b = 0; b < B_BATCH; ++b) {
      ps += bp[2 * b + 0];
      pl += bp[2 * b + 1];
    }
    out[0] = (ps / (float)B_BATCH) * 1.0f;  // PUSH_W
    out[1] = (pl / (float)B_BATCH) * 1.0f;  // PULL_W
  }
}

extern "C" void kernel_launch(void* const* d_in, const int* in_sizes, int n_in,
                              void* d_out, int out_size, void* d_ws, size_t ws_size,
                              hipStream_t stream) {
  // setup_inputs order: gt_inds, anchor_gt_inds, gt_bboxes, proposal_list
  const int*   gti  = (const int*)d_in[1];    // anchor_gt_inds (B,N)
  const float* gtb  = (const float*)d_in[2];  // gt_bboxes      (B,G,4)
  const float* prop = (const float*)d_in[3];  // proposal_list  (B,N,5)
  float* ws  = (float*)d_ws;                  // (B,2) per-batch push,pull
  float* out = (float*)d_out;                 // (2,)

  nms_loss_batch_kernel<<<B_BATCH, NTHREADS, 0, stream>>>(gti, gtb, prop, ws);
  nms_loss_finalize_kernel<<<1, 32, 0, stream>>>(ws, out);
}